// TokenEmbedding_40132174413951
// MI455X (gfx1250) — compile-verified
//
#include <hip/hip_runtime.h>
#include <hip/hip_bf16.h>

// Problem constants from the reference: B=16, L=4096, H=768, f32 data,
// sorted int32 segment ids per batch row.
#define BN 16
#define LN 4096
#define HN 768
#define TOK_TILES (LN / 16)      // 256 token tiles per batch
#define COL_TILES (HN / 16)      // 48 column tiles
#define WAVES_PER_WG 8
#define TILES_PER_WAVE (COL_TILES / WAVES_PER_WG) // 6

typedef __attribute__((ext_vector_type(2))) float v2f;
typedef __attribute__((ext_vector_type(8))) float v8f;

__device__ __forceinline__ int lower_bound(const int* __restrict__ a,
                                           int lo, int hi, int val) {
    // first index i in [lo,hi) with a[i] >= val
    while (lo < hi) {
        int mid = (lo + hi) >> 1;
        if (a[mid] < val) lo = mid + 1; else hi = mid;
    }
    return lo;
}

// One workgroup per (batch, 16-token tile). D = S * X over the contiguous
// wordpiece range feeding these 16 tokens, done with f32 WMMA (exact 0/1 A).
__global__ __launch_bounds__(256)
void segment_sum_wmma_kernel(const float* __restrict__ x,     // [B, L, H]
                             const int* __restrict__ seg,     // [B, L] sorted
                             float* __restrict__ out)         // [B, L, H]
{
    const int wg    = blockIdx.x;
    const int b     = wg >> 8;            // / TOK_TILES
    const int ttile = wg & (TOK_TILES - 1);
    const int j0    = ttile * 16;

    const int* __restrict__ segb = seg + b * LN;

    // Contributing wordpieces are exactly seg in [j0, j0+16) -> contiguous [s,e)
    const int s = lower_bound(segb, 0, LN, j0);
    const int e = lower_bound(segb, s, LN, j0 + 16);

    const int tid  = threadIdx.x;
    const int lane = tid & 31;
    const int wave = tid >> 5;            // 0..7
    const int n    = lane & 15;           // N (cols) / M (rows of A) index
    const int half = lane >> 4;           // 0 or 1
    const int kh   = half << 1;           // K offset of this lane-half

    const float* __restrict__ xb = x + (size_t)b * LN * HN;
    float* __restrict__ ob       = out + (size_t)b * LN * HN;

    v8f acc[TILES_PER_WAVE];
#pragma unroll
    for (int t = 0; t < TILES_PER_WAVE; ++t) acc[t] = (v8f){0.f,0.f,0.f,0.f,0.f,0.f,0.f,0.f};

    const int colbase = wave * TILES_PER_WAVE * 16;

    for (int kb = s; kb < e; kb += 4) {
        // A/B K layout (ISA 7.12.2): K = vgpr + 2*(lane>>4)
        const int p0  = kb + kh;
        const int p1  = p0 + 1;
        const int p0c = p0 < LN ? p0 : LN - 1;   // clamp; A=0 kills OOB cols
        const int p1c = p1 < LN ? p1 : LN - 1;

        // Unconditional loads + bitwise predicate combine: no divergent
        // branches in the K-loop, EXEC stays all-ones for the WMMAs.
        const int s0v = segb[p0c];
        const int s1v = segb[p1c];
        const int m0 = (int)(p0 < e) & (int)(s0v == j0 + n);
        const int m1 = (int)(p1 < e) & (int)(s1v == j0 + n);
        v2f a;
        a.x = m0 ? 1.0f : 0.0f;
        a.y = m1 ? 1.0f : 0.0f;

        const float* __restrict__ r0 = xb + (size_t)p0c * HN;
        const float* __restrict__ r1 = xb + (size_t)p1c * HN;

#pragma unroll
        for (int t = 0; t < TILES_PER_WAVE; ++t) {
            const int col = colbase + t * 16 + n;
            v2f bb;
            bb.x = r0[col];
            bb.y = r1[col];
            // D = A(16x4, 0/1 selection) * B(4x16, data) + C  -- exact f32
            acc[t] = __builtin_amdgcn_wmma_f32_16x16x4_f32(
                /*neg_a=*/false, a, /*neg_b=*/false, bb,
                /*c_mod=*/(short)0, acc[t],
                /*reuse_a=*/false, /*reuse_b=*/false);
        }
    }

    // C/D layout: VGPR r -> row (j0 + r + 8*half), col = colbase + t*16 + n.
    // Every (b, token, col) is written by exactly one workgroup; empty tokens
    // get zeros from the untouched accumulator.
#pragma unroll
    for (int t = 0; t < TILES_PER_WAVE; ++t) {
        const int col = colbase + t * 16 + n;
        const int rowbase = j0 + half * 8;
#pragma unroll
        for (int r = 0; r < 8; ++r) {
            ob[(size_t)(rowbase + r) * HN + col] = acc[t][r];
        }
    }
}

extern "C" void kernel_launch(void* const* d_in, const int* in_sizes, int n_in,
                              void* d_out, int out_size, void* d_ws, size_t ws_size,
                              hipStream_t stream) {
    const float* x   = (const float*)d_in[0];   // sequence_output [16,4096,768] f32
    const int*   seg = (const int*)d_in[1];     // wordpiece_to_token [16,4096] i32
    float*       out = (float*)d_out;           // [16,4096,768] f32

    dim3 grid(BN * TOK_TILES);  // 4096 workgroups
    dim3 block(256);            // 8 wave32s
    segment_sum_wmma_kernel<<<grid, block, 0, stream>>>(x, seg, out);
}